// ApplyDF_29231547416739
// MI455X (gfx1250) — compile-verified
//
#include <hip/hip_runtime.h>
#include <cstdint>

// ---------------------------------------------------------------------------
// DeepFilterNet apply_df for MI455X (gfx1250).
// Memory-bound (~738 MB min traffic @ 23.3 TB/s => ~32 us). No GEMM structure
// (per-element coefficients) => WMMA inapplicable; CDNA5 feature used is the
// async global<->LDS mover for the 394 MB pass-through copy.
// ---------------------------------------------------------------------------

#define DF_ORDER 5
#define DF_NB    96
#define DF_T     2000
#define DF_F     481
#define DF_B     32
#define CT       40                 // t-chunk per block
#define NCHUNK   (DF_T / CT)        // 50

// ---------------------- filter kernel (f < 96) -----------------------------
__global__ __launch_bounds__(DF_NB) void df_apply_kernel(
    const float* __restrict__ spec, const float* __restrict__ coefs,
    float* __restrict__ out)
{
    const int f  = threadIdx.x;      // 0..95
    const int b  = blockIdx.x;       // 0..31
    const int t0 = blockIdx.y * CT;  // chunk start

    // sliding window: w[n] holds spec at time (t + n - 4)
    float2 w[DF_ORDER];
#pragma unroll
    for (int k = 0; k < DF_ORDER - 1; ++k) {
        int t = t0 - (DF_ORDER - 1) + k;
        if (t >= 0)
            w[k] = *(const float2*)(spec + 2 * (((size_t)b * DF_T + t) * DF_F + f));
        else
            w[k] = make_float2(0.f, 0.f);
    }

    const float* sp = spec  + 2 * (((size_t)b * DF_T + t0) * DF_F + f);
    float*       op = out   + 2 * (((size_t)b * DF_T + t0) * DF_F + f);
    const float* cp = coefs + 2 * ((((size_t)b * DF_ORDER) * DF_T + t0) * DF_NB + f);

    const size_t srow = (size_t)DF_F * 2;            // spec/out t-stride (floats)
    const size_t ctap = (size_t)DF_T * DF_NB * 2;    // coef tap-stride
    const size_t crow = (size_t)DF_NB * 2;           // coef t-stride

#pragma unroll 2
    for (int tt = 0; tt < CT; ++tt) {
        w[DF_ORDER - 1] = *(const float2*)sp;
        float ar = 0.f, ai = 0.f;
#pragma unroll
        for (int n = 0; n < DF_ORDER; ++n) {
            float2 c = *(const float2*)(cp + (size_t)n * ctap);
            ar = fmaf(w[n].x,  c.x, ar);
            ar = fmaf(-w[n].y, c.y, ar);
            ai = fmaf(w[n].x,  c.y, ai);
            ai = fmaf(w[n].y,  c.x, ai);
        }
        *(float2*)op = make_float2(ar, ai);
#pragma unroll
        for (int k = 0; k < DF_ORDER - 1; ++k) w[k] = w[k + 1];
        sp += srow; op += srow; cp += crow;
    }
}

// ---------------------- pass-through copy (f >= 96) ------------------------
#define CPB 256                     // threads per block
#define CPU 8                       // float2 per thread per phase
#define PHASE_ELEMS (CPB * CPU)     // 2048 float2 = 16 KB per phase
#define ROW2 385                    // float2 per row in the copy region
#define NCPY2 ((unsigned)(DF_B * DF_T) * (unsigned)ROW2)   // 24,640,000

#if __has_builtin(__builtin_amdgcn_global_load_async_to_lds_b64) && \
    __has_builtin(__builtin_amdgcn_global_store_async_from_lds_b64)
#define USE_ASYNC_COPY 1
#else
#define USE_ASYNC_COPY 0
#endif

#if USE_ASYNC_COPY
#if __has_builtin(__builtin_amdgcn_s_wait_asynccnt)
#define WAIT_ASYNC0() __builtin_amdgcn_s_wait_asynccnt(0)
#else
#define WAIT_ASYNC0() asm volatile("s_wait_asynccnt 0x0" ::: "memory")
#endif
// Builtin signature (from hipcc diagnostic): (int2 AS1*, int2 AS3*, Ii, Ii)
typedef int v2i __attribute__((ext_vector_type(2)));
typedef __attribute__((address_space(1))) v2i* g_b64_t;
typedef __attribute__((address_space(3))) v2i* l_b64_t;
#endif

__global__ __launch_bounds__(CPB) void df_copy_kernel(
    const float* __restrict__ spec, float* __restrict__ out)
{
    const unsigned tid    = threadIdx.x;
    const unsigned nphase = (NCPY2 + PHASE_ELEMS - 1) / PHASE_ELEMS;

#if USE_ASYNC_COPY
    __shared__ float2 buf[2][PHASE_ELEMS];   // 32 KB (of 320 KB/WGP)
    int s = 0;
    unsigned phase = blockIdx.x;

    if (phase < nphase) {
        // prime: async-gather phase 0 tile into buf[0]
#pragma unroll
        for (int u = 0; u < CPU; ++u) {
            unsigned e = phase * PHASE_ELEMS + (unsigned)u * CPB + tid;
            if (e < NCPY2) {
                unsigned row = e / ROW2, col = e - row * ROW2;
                const float2* g = (const float2*)spec + (size_t)row * DF_F + DF_NB + col;
                __builtin_amdgcn_global_load_async_to_lds_b64(
                    (g_b64_t)g, (l_b64_t)&buf[0][u * CPB + tid], 0, 0);
            }
        }
    }

    for (; phase < nphase; phase += gridDim.x) {
        unsigned next = phase + gridDim.x;
        WAIT_ASYNC0();   // loads into buf[s] (and prior stores) complete

        if (next < nphase) {
            // overlap: async-load next tile into the other buffer
#pragma unroll
            for (int u = 0; u < CPU; ++u) {
                unsigned e = next * PHASE_ELEMS + (unsigned)u * CPB + tid;
                if (e < NCPY2) {
                    unsigned row = e / ROW2, col = e - row * ROW2;
                    const float2* g = (const float2*)spec + (size_t)row * DF_F + DF_NB + col;
                    __builtin_amdgcn_global_load_async_to_lds_b64(
                        (g_b64_t)g, (l_b64_t)&buf[s ^ 1][u * CPB + tid], 0, 0);
                }
            }
        }
        // async-scatter current tile from buf[s] to out
#pragma unroll
        for (int u = 0; u < CPU; ++u) {
            unsigned e = phase * PHASE_ELEMS + (unsigned)u * CPB + tid;
            if (e < NCPY2) {
                unsigned row = e / ROW2, col = e - row * ROW2;
                float2* g = (float2*)out + (size_t)row * DF_F + DF_NB + col;
                __builtin_amdgcn_global_store_async_from_lds_b64(
                    (g_b64_t)g, (l_b64_t)&buf[s][u * CPB + tid], 0, 0);
            }
        }
        s ^= 1;
    }
    // outstanding async stores drain at s_endpgm (implicit wait-idle)
#else
    for (unsigned phase = blockIdx.x; phase < nphase; phase += gridDim.x) {
#pragma unroll
        for (int u = 0; u < CPU; ++u) {
            unsigned e = phase * PHASE_ELEMS + (unsigned)u * CPB + tid;
            if (e < NCPY2) {
                unsigned row = e / ROW2, col = e - row * ROW2;
                size_t idx = (size_t)row * DF_F + DF_NB + col;
                ((float2*)out)[idx] = ((const float2*)spec)[idx];
            }
        }
    }
#endif
}

// ---------------------------------------------------------------------------
extern "C" void kernel_launch(void* const* d_in, const int* in_sizes, int n_in,
                              void* d_out, int out_size, void* d_ws, size_t ws_size,
                              hipStream_t stream)
{
    const float* spec  = (const float*)d_in[0];
    const float* coefs = (const float*)d_in[1];
    float*       out   = (float*)d_out;

    dim3 gridc(DF_B, NCHUNK);
    df_apply_kernel<<<gridc, DF_NB, 0, stream>>>(spec, coefs, out);
    df_copy_kernel<<<4096, CPB, 0, stream>>>(spec, out);

    (void)in_sizes; (void)n_in; (void)out_size; (void)d_ws; (void)ws_size;
}